// SeesawLoss_81647328297606
// MI455X (gfx1250) — compile-verified
//
#include <hip/hip_runtime.h>
#include <hip/hip_bf16.h>
#include <math.h>

#ifndef __has_builtin
#define __has_builtin(x) 0
#endif

// ---------------- problem constants (fixed by the reference harness) --------
#define C_CLS      1203
#define NCOL       1205                    // C + 2
#define GROUP_ROWS 4
#define CHUNK_FLOATS (GROUP_ROWS * NCOL)   // 4820 floats = 19280 bytes (16B multiple)
#define CHUNK_SEGS   (CHUNK_FLOATS / 4)    // 1205 x 16-byte segments
#define BLOCK      128                     // 4 waves -> one wave per row in the group
#define SEG_ITERS  ((CHUNK_SEGS + BLOCK - 1) / BLOCK)   // 10 b128 async ops per wave/chunk
#define KPL        38                      // ceil(1203 / 32) columns per lane

#define P_POW   0.8f
#define Q_POW   2.0f
#define LOG_EPS (-4.6051701859880914f)     // log(0.01)

// ---------------- CDNA5 async global->LDS path ------------------------------
typedef int v4i __attribute__((ext_vector_type(4)));
typedef __attribute__((address_space(1))) v4i  gv4i;
typedef __attribute__((address_space(3))) v4i  lv4i;
typedef __attribute__((address_space(1))) char g1char;
typedef __attribute__((address_space(3))) char l3char;

__device__ __forceinline__ void async_load_chunk(const float* g,
                                                 float* l /*LDS*/, int tid) {
    g1char* gp = (g1char*)(const_cast<float*>(g));
    l3char* lp = (l3char*)l;
#pragma unroll
    for (int s = 0; s < SEG_ITERS; ++s) {
        int idx = tid + s * BLOCK;
        if (idx > CHUNK_SEGS - 1) idx = CHUNK_SEGS - 1;   // clamp: dup loads, same data
#if __has_builtin(__builtin_amdgcn_global_load_async_to_lds_b128)
        __builtin_amdgcn_global_load_async_to_lds_b128(
            (gv4i*)(gp + (size_t)idx * 16),
            (lv4i*)(lp + idx * 16), 0, 0);
#else
        unsigned lds_off = (unsigned)(size_t)(lp + idx * 16);
        unsigned g_off   = (unsigned)idx * 16u;
        asm volatile("global_load_async_to_lds_b128 %0, %1, %2 offset:0"
                     :: "v"(lds_off), "v"(g_off),
                        "s"((unsigned long long)(size_t)g)
                     : "memory");
#endif
    }
}

template <int NV>
__device__ __forceinline__ void wait_async() {
#if __has_builtin(__builtin_amdgcn_s_wait_asynccnt)
    __builtin_amdgcn_s_wait_asynccnt(NV);
#else
    asm volatile("s_wait_asynccnt %0" :: "n"(NV));
#endif
}

// ---------------- wave32 reductions -----------------------------------------
__device__ __forceinline__ float wredMax(float x) {
#pragma unroll
    for (int o = 16; o > 0; o >>= 1) x = fmaxf(x, __shfl_xor(x, o, 32));
    return x;
}
__device__ __forceinline__ float wredSum(float x) {
#pragma unroll
    for (int o = 16; o > 0; o >>= 1) x += __shfl_xor(x, o, 32);
    return x;
}
__device__ __forceinline__ float softplusf(float x) {
    return fmaxf(x, 0.0f) + log1pf(__expf(-fabsf(x)));
}

// ---------------- prep kernels ----------------------------------------------
__global__ void seesaw_init(int* __restrict__ hist, float* __restrict__ out) {
    int i = blockIdx.x * blockDim.x + threadIdx.x;
    if (i < C_CLS + 1) hist[i] = 0;
    if (i == 0) out[0] = 0.0f;
}
__global__ void seesaw_hist(const int* __restrict__ labels, int n,
                            int* __restrict__ hist) {
    int i = blockIdx.x * blockDim.x + threadIdx.x;
    if (i < n) atomicAdd(&hist[labels[i]], 1);
}
__global__ void seesaw_lcum(const int* __restrict__ hist, float* __restrict__ lcum) {
    int i = blockIdx.x * blockDim.x + threadIdx.x;
    if (i < C_CLS + 1) lcum[i] = __logf(fmaxf((float)hist[i], 1.0f));
}

// ---------------- main kernel: one wave per row, async double-buffer --------
__global__ void __launch_bounds__(BLOCK)
seesaw_main(const float* __restrict__ cls, const int* __restrict__ labels,
            const float* __restrict__ lweights, const int* __restrict__ avgp,
            const float* __restrict__ lcum, float* __restrict__ out, int n) {
    __shared__ __align__(16) float sbuf[2][CHUNK_FLOATS];
    __shared__ float slcum[C_CLS + 1];

    const int tid  = threadIdx.x;
    const int lane = tid & 31;
    const int w    = tid >> 5;      // wave index in block = row within group

    for (int i = tid; i < C_CLS + 1; i += BLOCK) slcum[i] = lcum[i];

    const float inv_avg = 1.0f / (float)(*avgp);
    const int groups = n / GROUP_ROWS;
    const int gs = gridDim.x;
    int g = blockIdx.x;
    int cur = 0;

    if (g < groups) async_load_chunk(cls + (size_t)g * CHUNK_FLOATS, sbuf[0], tid);
    __syncthreads();   // slcum visible to all waves

    while (g < groups) {
        const int gn = g + gs;
        if (gn < groups) {
            async_load_chunk(cls + (size_t)gn * CHUNK_FLOATS, sbuf[cur ^ 1], tid);
            wait_async<SEG_ITERS>();     // current chunk complete; next in flight
        } else {
            wait_async<0>();
        }
        __syncthreads();                 // all waves' slices landed in LDS

        {
            const float* rp = &sbuf[cur][w * NCOL];
            const int row = g * GROUP_ROWS + w;

            // Pass 1: row max over class logits (register-resident row)
            float v[KPL];
            float m = -INFINITY;
#pragma unroll
            for (int k = 0; k < KPL; ++k) {
                const int j = lane + 32 * k;
                v[k] = (j < C_CLS) ? rp[j] : -INFINITY;
                m = fmaxf(m, v[k]);
            }
            m = wredMax(m);

            // Pass 2: log-sum-exp denominator
            float s = 0.0f;
#pragma unroll
            for (int k = 0; k < KPL; ++k) s += __expf(v[k] - m);
            s = wredSum(s);
            const float L = m + __logf(s);           // log-denominator

            const int   lab  = labels[row];
            const float vlab = rp[lab];              // LDS broadcast read
            const float lcl  = slcum[lab];
            const float t    = fmaxf(vlab - L, LOG_EPS);  // log(max(self, EPS))

            // Pass 3: adjusted logits, excluded at the label column
            float a[KPL];
            float amax = -INFINITY;
#pragma unroll
            for (int k = 0; k < KPL; ++k) {
                const int j = lane + 32 * k;
                float aj = -INFINITY;
                if (j < C_CLS && j != lab) {
                    aj = v[k] + P_POW * fminf(slcum[j] - lcl, 0.0f)
                              + Q_POW * fmaxf(v[k] - L - t, 0.0f);
                }
                a[k] = aj;
                amax = fmaxf(amax, aj);
            }
            amax = wredMax(amax);

            // Pass 4: neg logsumexp
            float ss = 0.0f;
#pragma unroll
            for (int k = 0; k < KPL; ++k) ss += __expf(a[k] - amax);
            ss = wredSum(ss);
            const float neg_lse = amax + __logf(ss);

            const float wgt = lweights[row];
            // masks select entries where mask*lw == 1.0 (reference semantics)
            const float cls_term = (wgt == 1.0f) ? softplusf(neg_lse - vlab) : 0.0f;
            // binary CE: labels<C so b_label==0 -> -log_softmax[...,0] = softplus(b1-b0)
            const float bin_term = softplusf(rp[C_CLS + 1] - rp[C_CLS]) * wgt;

            if (lane == 0) atomicAdd(out, (cls_term + bin_term) * inv_avg);
        }

        __syncthreads();   // everyone done reading sbuf[cur] before it is re-filled
        cur ^= 1;
        g = gn;
    }
}

// ---------------- host entry -------------------------------------------------
extern "C" void kernel_launch(void* const* d_in, const int* in_sizes, int n_in,
                              void* d_out, int out_size, void* d_ws, size_t ws_size,
                              hipStream_t stream) {
    const float* cls    = (const float*)d_in[0];
    const int*   labels = (const int*)d_in[1];
    const float* lw     = (const float*)d_in[2];
    const int*   avgp   = (const int*)d_in[3];
    float*       out    = (float*)d_out;

    const int n = in_sizes[1];   // number of rows

    int*   hist = (int*)d_ws;
    float* lcum = (float*)((char*)d_ws + 8192);

    seesaw_init<<<(C_CLS + 1 + 255) / 256, 256, 0, stream>>>(hist, out);
    seesaw_hist<<<(n + 255) / 256, 256, 0, stream>>>(labels, n, hist);
    seesaw_lcum<<<(C_CLS + 1 + 255) / 256, 256, 0, stream>>>(hist, lcum);

    seesaw_main<<<1024, BLOCK, 0, stream>>>(cls, labels, lw, avgp, lcum, out, n);
}